// CappedMean_87333864997204
// MI455X (gfx1250) — compile-verified
//
#include <hip/hip_runtime.h>
#include <hip/hip_bf16.h>

typedef __attribute__((ext_vector_type(2))) float v2f;
typedef __attribute__((ext_vector_type(8))) float v8f;

// CappedMean: out[b, d] = (1/N[b]) * sum_{r < N[b]} x[b, r, d]
//
// One block per batch; one wave32 per 16-column tile (D/16 waves per block).
// The row reduction runs on the WMMA pipe: D = ones(16x4) x B(4x16) + C, so
// each v_wmma_f32_16x16x4_f32 folds 4 rows x 16 cols into the C accumulator.
// 16 rows per loop iteration across 4 independent accumulators -> 8 NT loads
// in flight per lane, one 32-bit offset bump per iteration (GVS addressing,
// row deltas folded into the 24-bit immediate offset when D is constexpr).
//
// DD is the compile-time D (0 = use runtime Drt).
template <int DD>
__global__ __launch_bounds__(1024)
void capped_mean_wmma_kernel(const float* __restrict__ x,
                             const int* __restrict__ N,
                             float* __restrict__ out,
                             int L, int Drt) {
  const int D    = DD ? DD : Drt;
  const int b    = blockIdx.x;
  const int wave = threadIdx.x >> 5;          // 0..(D/16 - 1)
  const int lane = threadIdx.x & 31;
  const int col  = (wave << 4) + (lane & 15);
  // lanes 0-15 supply two K-rows, lanes 16-31 the other two; the exact k
  // assignment is irrelevant (A is all-ones, WMMA sums every B element per col)
  const int rowoff = (lane >> 4) << 1;        // 0 or 2

  const float* __restrict__ xb = x + (size_t)b * (size_t)L * (size_t)D;
  const int n = N[b];                          // block-uniform -> no divergence

  v2f a; a.x = 1.0f; a.y = 1.0f;              // ones A-matrix (16x4)
  v8f c0 = {}, c1 = {}, c2 = {}, c3 = {};     // 4 independent C/D accumulators

  unsigned off = (unsigned)(rowoff * D + col); // 32-bit element offset
  int r = 0;
  const int nfull = n & ~15;                   // full 16-row chunks
  for (; r < nfull; r += 16) {
    v2f b0, b1, b2, b3;
    b0.x = __builtin_nontemporal_load(xb + off);            // rows r+0..3
    b0.y = __builtin_nontemporal_load(xb + off + 1 * D);
    b1.x = __builtin_nontemporal_load(xb + off + 4 * D);    // rows r+4..7
    b1.y = __builtin_nontemporal_load(xb + off + 5 * D);
    b2.x = __builtin_nontemporal_load(xb + off + 8 * D);    // rows r+8..11
    b2.y = __builtin_nontemporal_load(xb + off + 9 * D);
    b3.x = __builtin_nontemporal_load(xb + off + 12 * D);   // rows r+12..15
    b3.y = __builtin_nontemporal_load(xb + off + 13 * D);
    // speculative prefetch one chunk (16 KB) ahead -> global_prefetch_b8
    __builtin_prefetch(xb + off + 16 * D, 0, 0);
    c0 = __builtin_amdgcn_wmma_f32_16x16x4_f32(false, a, false, b0,
                                               (short)0, c0, false, false);
    c1 = __builtin_amdgcn_wmma_f32_16x16x4_f32(false, a, false, b1,
                                               (short)0, c1, false, false);
    c2 = __builtin_amdgcn_wmma_f32_16x16x4_f32(false, a, false, b2,
                                               (short)0, c2, false, false);
    c3 = __builtin_amdgcn_wmma_f32_16x16x4_f32(false, a, false, b3,
                                               (short)0, c3, false, false);
    off += 16u * (unsigned)D;
  }

  // Masked tail: up to 15 rows, one predicated 16-row chunk. Out-of-range
  // rows contribute zeros; EXEC reconverges before the WMMAs (which require
  // EXEC == all-1s). Runs once, so the serialized cmpx blocks don't matter.
  if (r < n) {
    const int r0 = r + rowoff;
    v2f b0, b1, b2, b3;
    b0.x = (r0 +  0 < n) ? xb[off]          : 0.0f;
    b0.y = (r0 +  1 < n) ? xb[off + 1 * D]  : 0.0f;
    b1.x = (r0 +  4 < n) ? xb[off + 4 * D]  : 0.0f;
    b1.y = (r0 +  5 < n) ? xb[off + 5 * D]  : 0.0f;
    b2.x = (r0 +  8 < n) ? xb[off + 8 * D]  : 0.0f;
    b2.y = (r0 +  9 < n) ? xb[off + 9 * D]  : 0.0f;
    b3.x = (r0 + 12 < n) ? xb[off + 12 * D] : 0.0f;
    b3.y = (r0 + 13 < n) ? xb[off + 13 * D] : 0.0f;
    c0 = __builtin_amdgcn_wmma_f32_16x16x4_f32(false, a, false, b0,
                                               (short)0, c0, false, false);
    c1 = __builtin_amdgcn_wmma_f32_16x16x4_f32(false, a, false, b1,
                                               (short)0, c1, false, false);
    c2 = __builtin_amdgcn_wmma_f32_16x16x4_f32(false, a, false, b2,
                                               (short)0, c2, false, false);
    c3 = __builtin_amdgcn_wmma_f32_16x16x4_f32(false, a, false, b3,
                                               (short)0, c3, false, false);
  }

  // C/D layout: VGPR0, lanes 0-15 hold (M=0, N=lane); all M rows identical
  // (A is all-ones), so lane<16 / element 0 carries the column sums.
  if (lane < 16) {
    const float s = (c0[0] + c1[0]) + (c2[0] + c3[0]);
    out[(size_t)b * D + col] = s * (1.0f / (float)n);
  }
}

extern "C" void kernel_launch(void* const* d_in, const int* in_sizes, int n_in,
                              void* d_out, int out_size, void* d_ws, size_t ws_size,
                              hipStream_t stream) {
  const float* x = (const float*)d_in[0];
  const int*   N = (const int*)d_in[1];
  float*     out = (float*)d_out;

  const int B = in_sizes[1];            // 2048
  const int D = out_size / B;           // 256
  const int L = in_sizes[0] / (B * D);  // 512

  const dim3 grid(B);
  const dim3 block((D / 16) * 32);      // one wave32 per 16-column tile

  if (D == 256) {
    capped_mean_wmma_kernel<256><<<grid, block, 0, stream>>>(x, N, out, L, D);
  } else {
    capped_mean_wmma_kernel<0><<<grid, block, 0, stream>>>(x, N, out, L, D);
  }
}